// InvariantMessage_1451698946759
// MI455X (gfx1250) — compile-verified
//
#include <hip/hip_runtime.h>

// ---------------------------------------------------------------------------
// PaiNN invariant-message kernel for gfx1250 (MI455X), fp32 WMMA path.
//
// inv = silu(s_j@W1+b1)@W2 + b2            (nodes, staged in d_ws, 30.7 MB)
// out = inv[nbrs[:,1]] * ((rbf@Wr+br)*env) (edges, 491 MB -> HBM-bound)
//
// Matrix ops use V_WMMA_F32_16X16X4_F32 (wave32, 16x16 f32 tiles, K=4/instr).
// ---------------------------------------------------------------------------

typedef __attribute__((ext_vector_type(2))) float v2f;
typedef __attribute__((ext_vector_type(8))) float v8f;

#define FEAT        128
#define THREE_FEAT  384
#define NRBF        20
#define CUTOFF_F    5.0f
#define PI_F        3.14159265358979323846f

__device__ __forceinline__ v8f wmma_f32_k4(v2f a, v2f b, v8f c) {
  // 8 args: (neg_a, A, neg_b, B, c_mod, C, reuse_a, reuse_b)
  return __builtin_amdgcn_wmma_f32_16x16x4_f32(
      false, a, false, b, (short)0, c, false, false);
}

// ---------------------------------------------------------------------------
// Kernel 1: per 16-node tile, fused two-layer MLP.
// Block = 128 threads = 4 waves. Wave w handles N-tiles w, w+4, ...
// GEMM1 result (h, 16x128) staged in LDS; GEMM2 writes inv to workspace.
// ---------------------------------------------------------------------------
__global__ __launch_bounds__(128)
void node_mlp_kernel(const float* __restrict__ s_j,
                     const float* __restrict__ W1,
                     const float* __restrict__ b1,
                     const float* __restrict__ W2,
                     const float* __restrict__ b2,
                     float* __restrict__ inv_ws,
                     int n_nodes) {
  __shared__ float h_s[16 * FEAT];  // 8 KB

  const int wave = threadIdx.x >> 5;
  const int lane = threadIdx.x & 31;
  const int m    = lane & 15;   // row within 16-tile / col within 16-tile
  const int half = lane >> 4;   // lane-half select (K/M split per ISA layout)
  const int m0   = blockIdx.x * 16;

  int arow = m0 + m;
  if (arow >= n_nodes) arow = n_nodes - 1;  // clamp (tail tile)

  // ---- GEMM1: h = silu(s_j @ W1 + b1); 8 N-tiles split over 4 waves ----
  for (int nt = wave; nt < FEAT / 16; nt += 4) {
    const int col = nt * 16 + m;
    v8f c = {};
    for (int k = 0; k < FEAT; k += 4) {
      const int kk = k + 2 * half;
      v2f a, b;
      a.x = s_j[(size_t)arow * FEAT + kk];
      a.y = s_j[(size_t)arow * FEAT + kk + 1];
      b.x = W1[(size_t)kk * FEAT + col];
      b.y = W1[(size_t)(kk + 1) * FEAT + col];
      c = wmma_f32_k4(a, b, c);
    }
    const float bias = b1[col];
#pragma unroll
    for (int r = 0; r < 8; ++r) {
      const int rr = r + 8 * half;          // C/D layout: row = r + 8*half
      const float x = c[r] + bias;
      h_s[rr * FEAT + col] = x / (1.0f + __expf(-x));  // silu
    }
  }
  __syncthreads();

  // ---- GEMM2: inv = h @ W2 + b2; 24 N-tiles split over 4 waves ----
  for (int nt = wave; nt < THREE_FEAT / 16; nt += 4) {
    const int col = nt * 16 + m;
    v8f c = {};
    for (int k = 0; k < FEAT; k += 4) {
      const int kk = k + 2 * half;
      v2f a, b;
      a.x = h_s[m * FEAT + kk];
      a.y = h_s[m * FEAT + kk + 1];
      b.x = W2[(size_t)kk * THREE_FEAT + col];
      b.y = W2[(size_t)(kk + 1) * THREE_FEAT + col];
      c = wmma_f32_k4(a, b, c);
    }
    const float bias = b2[col];
#pragma unroll
    for (int r = 0; r < 8; ++r) {
      const int rr  = r + 8 * half;
      const int row = m0 + rr;
      if (row < n_nodes)
        inv_ws[(size_t)row * THREE_FEAT + col] = c[r] + bias;
    }
  }
}

// ---------------------------------------------------------------------------
// Kernel 2: per 16-edge tile (one wave per tile, 4 tiles per block).
// Builds rbf(16x20) in LDS, 5 A-fragments reused across 24 N-tiles of Wr,
// fused epilogue: +br, *envelope, *phi-gather (inv resident in L2), store.
// ---------------------------------------------------------------------------
__global__ __launch_bounds__(128)
void edge_kernel(const float* __restrict__ dist,
                 const int*   __restrict__ nbrs,
                 const float* __restrict__ Wr,
                 const float* __restrict__ br,
                 const float* __restrict__ inv_ws,
                 float* __restrict__ out,
                 int n_edges) {
  __shared__ float rbf_s[4][16 * NRBF];  // 5 KB
  __shared__ float env_s[4][16];
  __shared__ int   jidx_s[4][16];

  const int wave = threadIdx.x >> 5;
  const int lane = threadIdx.x & 31;
  const int m    = lane & 15;
  const int half = lane >> 4;
  const int e0   = (blockIdx.x * 4 + wave) * 16;

  // --- RBF generation: lane (m, half) computes 10 of 20 RBFs for edge e0+m ---
  int e = e0 + m;
  if (e >= n_edges) e = n_edges - 1;  // clamp; tail guarded at store
  const float d     = dist[e];
  const float inv_d = 1.0f / d;
  const float w0    = PI_F / CUTOFF_F;
#pragma unroll
  for (int t = 0; t < NRBF / 2; ++t) {
    const int n = half * (NRBF / 2) + t;
    rbf_s[wave][m * NRBF + n] = __sinf(d * ((float)(n + 1) * w0)) * inv_d;
  }
  if (half == 0) {
    const float env = 0.5f * (__cosf(d * w0) + 1.0f);
    env_s[wave][m]  = (d < CUTOFF_F) ? env : 0.0f;
    jidx_s[wave][m] = nbrs[2 * (size_t)e + 1];  // nbrs[:,1]
  }
  __syncthreads();

  // --- Hoist A-fragments (rbf) and per-row epilogue data into registers ---
  v2f afrag[5];
#pragma unroll
  for (int ks = 0; ks < 5; ++ks) {
    const int k = ks * 4 + 2 * half;            // max 19 == NRBF-1, no padding
    afrag[ks].x = rbf_s[wave][m * NRBF + k];
    afrag[ks].y = rbf_s[wave][m * NRBF + k + 1];
  }
  float env_r[8];
  int   j_r[8];
#pragma unroll
  for (int r = 0; r < 8; ++r) {
    const int rr = r + 8 * half;
    env_r[r] = env_s[wave][rr];
    j_r[r]   = jidx_s[wave][rr];
  }

  // --- 24 N-tiles: w_s = rbf @ Wr (+br), fused phi-gather & store ---
  for (int nt = 0; nt < THREE_FEAT / 16; ++nt) {
    const int col = nt * 16 + m;
    v8f c = {};
#pragma unroll
    for (int ks = 0; ks < 5; ++ks) {
      const int k = ks * 4 + 2 * half;
      v2f b;
      b.x = Wr[(size_t)k * THREE_FEAT + col];
      b.y = Wr[(size_t)(k + 1) * THREE_FEAT + col];
      c = wmma_f32_k4(afrag[ks], b, c);
    }
    const float bias = br[col];
#pragma unroll
    for (int r = 0; r < 8; ++r) {
      const int rr = r + 8 * half;
      const int eo = e0 + rr;
      if (eo < n_edges) {
        const float ws  = (c[r] + bias) * env_r[r];
        const float phi = inv_ws[(size_t)j_r[r] * THREE_FEAT + col];
        out[(size_t)eo * THREE_FEAT + col] = ws * phi;
      }
    }
  }
}

// ---------------------------------------------------------------------------
extern "C" void kernel_launch(void* const* d_in, const int* in_sizes, int n_in,
                              void* d_out, int out_size, void* d_ws, size_t ws_size,
                              hipStream_t stream) {
  const float* s_j  = (const float*)d_in[0];
  const float* dist = (const float*)d_in[1];
  const int*   nbrs = (const int*)d_in[2];
  const float* W1   = (const float*)d_in[3];
  const float* b1   = (const float*)d_in[4];
  const float* W2   = (const float*)d_in[5];
  const float* b2   = (const float*)d_in[6];
  const float* Wr   = (const float*)d_in[7];
  const float* br   = (const float*)d_in[8];
  float*       out  = (float*)d_out;
  float*       inv_ws = (float*)d_ws;  // n_nodes * 384 floats (30.7 MB)

  const int n_nodes = in_sizes[0] / FEAT;
  const int n_edges = in_sizes[1];

  const int node_blocks = (n_nodes + 15) / 16;
  node_mlp_kernel<<<node_blocks, 128, 0, stream>>>(s_j, W1, b1, W2, b2,
                                                   inv_ws, n_nodes);

  const int edge_tiles  = (n_edges + 15) / 16;
  const int edge_blocks = (edge_tiles + 3) / 4;
  edge_kernel<<<edge_blocks, 128, 0, stream>>>(dist, nbrs, Wr, br,
                                               inv_ws, out, n_edges);
}